// PDEConstraints_65498251264127
// MI455X (gfx1250) — compile-verified
//
#include <hip/hip_runtime.h>
#include <math.h>

#define SEQ_N   4096
#define BATCH_N 512

typedef __attribute__((ext_vector_type(4))) unsigned int v4u;
typedef __attribute__((ext_vector_type(4))) int          v4i;
typedef __attribute__((ext_vector_type(8))) int          v8i;

// --- CDNA5 TDM availability (guarded so host pass / other toolchains compile) ---
#if defined(__has_builtin)
#  if __has_builtin(__builtin_amdgcn_tensor_load_to_lds) && __has_builtin(__builtin_amdgcn_s_wait_tensorcnt)
#    define USE_TDM 1
#  endif
#endif
#ifndef USE_TDM
#  define USE_TDM 0
#endif

#if defined(__has_include)
#  if __has_include(<hip/amd_detail/amd_gfx1250_TDM.h>)
#    define TDM_6ARG 1   // therock-10.0 headers -> 6-arg builtin
#  endif
#endif
#ifndef TDM_6ARG
#  define TDM_6ARG 0     // ROCm 7.2 -> 5-arg builtin
#endif

__device__ __forceinline__ float frcp(float x) { return __builtin_amdgcn_rcpf(x); }

// Evaluate the 5 PDE residuals for one (s,b) point.
// model_output fields: q(unused here), Di, b, Dinf, n, T ; qi = q[s=0,b]
// Uses hardware v_exp/v_log/v_rcp; mirrors the reference's own f32
// pow(x,y)=exp(y*log(x)) structure (XLA lowers pow the same way).
__device__ __forceinline__ void pde_eval(float t, float Di, float b, float Dinf,
                                         float n, float T, float qi, float* o) {
    const float eps  = 1e-6f;
    const float Diqi = Di * qi;
    const float Dit  = Di * t;

    // pde1: exponential decline
    o[0] = Diqi * __expf(-Dit);
    // pde2: harmonic decline
    o[1] = Diqi * frcp(1.0f + Dit);
    // pde3: hyperbolic: Di*qi / max(1+b*Di*t, eps)^(1/b-1)
    {
        const float L = __logf(fmaxf(fmaf(b, Dit, 1.0f), eps));
        o[2] = Diqi * __expf((1.0f - frcp(b)) * L);
    }
    // pde4: Weibull: n*qi/Ts * (t/Ts)^(n-1) * exp(-(t/Ts)^n)
    {
        const float Ts  = fmaxf(T, eps);
        const float rTs = frcp(Ts);
        const float lx  = __logf(t * rTs);
        const float p   = __expf(n * lx);                 // (t/Ts)^n
        o[3] = (n * qi * rTs) * __expf((n - 1.0f) * lx) * __expf(-p);
    }
    // pde5: power-law exponential
    {
        const float ts    = fmaxf(t, eps);
        const float lt    = __logf(ts);
        const float term1 = fmaf(Di, __expf(-n * lt), -Dinf);   // Di*ts^-n - Dinf
        const float om    = 1.0f - n;
        // Di/max(eps,1-n) * (ts^(1-n) - 1) - Dinf*t   (same f32 cancellation as reference)
        const float ex    = fmaf(Di * frcp(fmaxf(eps, om)),
                                 __expf(om * lt) - 1.0f, -Dinf * t);
        o[4] = -(qi * term1 * __expf(ex));
    }
}

__global__ __launch_bounds__(256)
void PDEConstraints_65498251264127_kernel(const float* __restrict__ mo,
                                          const float* __restrict__ tin,
                                          float* __restrict__ out) {
    // Compacted q[0, :] column (512 floats, 2 KB) gathered into LDS.
    __shared__ __align__(16) float qi_row[BATCH_N];

    const int s   = blockIdx.x;    // one block per seq step
    const int tid = threadIdx.x;   // 256 threads, 2 batch elements each

#if USE_TDM
    if (tid < 32) {  // one wave drives the Tensor Data Mover
        const unsigned long long ga = (unsigned long long)(const void*)mo;
        // LDS byte offset of qi_row: low 32 bits of the generic pointer
        // (LDS aperture maps addr[31:0] to the LDS offset). Also escapes
        // qi_row so its loads can't be folded to undef.
        const unsigned lds_off = (unsigned)(unsigned long long)(&qi_row[0]);
        v4u g0;
        g0[0] = 1u;                                          // count=1 (valid D#)
        g0[1] = lds_off;                                     // lds_addr (bytes)
        g0[2] = (unsigned)(ga & 0xFFFFFFFFu);                // global_addr[31:0]
        g0[3] = (unsigned)((ga >> 32) & 0x01FFFFFFu)         // global_addr[56:32]
              | (2u << 30);                                  // type = 2 ("image")
        // Strided gather: 512 "rows" of 1 dword, row stride 6 dwords
        // -> pulls the q column of model_output[0,:,:], compacted in LDS.
        v8i g1;
        g1[0] = (int)(2u << 16);        // workgroup_mask=0, data_size=2 (4 bytes)
        g1[1] = (int)(6u << 16);        // tensor_dim0[15:0]=6 in bits[63:48]
        g1[2] = (int)(0x200u << 16);    // tensor_dim0[31:16]=0, tensor_dim1[15:0]=512
        g1[3] = (int)(1u << 16);        // tensor_dim1[31:16]=0, tile_dim0=1
        g1[4] = 512;                    // tile_dim1=512, tile_dim2=0
        g1[5] = 6;                      // tensor_dim0_stride = 6 dwords
        g1[6] = 0;
        g1[7] = 0;
        const v4i z = {0, 0, 0, 0};
#if TDM_6ARG
        const v8i z8 = {0, 0, 0, 0, 0, 0, 0, 0};
        __builtin_amdgcn_tensor_load_to_lds(g0, g1, z, z, z8, 0);
#else
        __builtin_amdgcn_tensor_load_to_lds(g0, g1, z, z, 0);
#endif
        __builtin_amdgcn_s_wait_tensorcnt(0);
    }
    // The TDM wrote LDS behind the compiler's back: treat qi_row as
    // potentially modified so the ds_loads below are not folded away.
    asm volatile("" : : "s"(&qi_row[0]) : "memory");
#else
    {   // fallback: cooperative strided gather of the q column
        for (int i = tid; i < BATCH_N; i += 256)
            qi_row[i] = mo[6 * i];
    }
#endif
    __syncthreads();

    // Two consecutive elements: (s, 2*tid) and (s, 2*tid+1).
    // 12 param floats = 3 aligned float4 loads (global_load_b128).
    const float4* mo4 = (const float4*)mo + (s * (BATCH_N * 6 / 4) + 3 * tid);
    const float4 A = mo4[0];   // q0  Di0 b0   Dinf0
    const float4 B = mo4[1];   // n0  T0  q1   Di1
    const float4 C = mo4[2];   // b1  Dinf1 n1 T1

    const float2 tt = ((const float2*)tin)[s * (BATCH_N / 2) + tid];

    // Conflict-free ds_load_b64: lane t reads LDS words 2t, 2t+1.
    const float2 qq = *(const float2*)&qi_row[2 * tid];

    float o0[5], o1[5];
    pde_eval(tt.x, A.y, A.z, A.w, B.x, B.y, qq.x, o0);
    pde_eval(tt.y, B.w, C.x, C.y, C.z, C.w, qq.y, o1);

    // 10 contiguous output floats -> five aligned float2 stores.
    float2* o2 = (float2*)out + (s * (BATCH_N * 5 / 2) + 5 * tid);
    o2[0] = make_float2(o0[0], o0[1]);
    o2[1] = make_float2(o0[2], o0[3]);
    o2[2] = make_float2(o0[4], o1[0]);
    o2[3] = make_float2(o1[1], o1[2]);
    o2[4] = make_float2(o1[3], o1[4]);
}

extern "C" void kernel_launch(void* const* d_in, const int* in_sizes, int n_in,
                              void* d_out, int out_size, void* d_ws, size_t ws_size,
                              hipStream_t stream) {
    const float* mo  = (const float*)d_in[0];   // [4096, 512, 6] f32
    const float* tin = (const float*)d_in[1];   // [4096, 512, 1] f32
    float*       out = (float*)d_out;           // [4096, 512, 5] f32
    (void)in_sizes; (void)n_in; (void)out_size; (void)d_ws; (void)ws_size;

    dim3 grid(SEQ_N);
    dim3 block(256);
    hipLaunchKernelGGL(PDEConstraints_65498251264127_kernel, grid, block, 0, stream,
                       mo, tin, out);
}